// CHGCNN_1202590843242
// MI455X (gfx1250) — compile-verified
//
#include <hip/hip_runtime.h>
#include <hip/hip_bf16.h>

// ---------------------------------------------------------------------------
// CHGCNN forward for MI455X (gfx1250, wave32).
// GEMMs via V_WMMA_F32_16X16X4_F32 (fp32-exact); scatter passes via f32
// atomics that stay resident in the 192MB L2.
// ---------------------------------------------------------------------------

#define N_NODES  100000
#define N_EDGES  400000
#define N_INC    1600000
#define N_GRAPHS 512
#define ATOM_DIM 92
#define NODE_DIM 64
#define H_DIM    128
#define NUM_LAYERS 3
#define BN_EPS   1e-5f

typedef float v2f __attribute__((ext_vector_type(2)));
typedef float v8f __attribute__((ext_vector_type(8)));

__device__ __forceinline__ float softplus_f(float x) {
  // stable: max(x,0) + log1p(exp(-|x|))
  return fmaxf(x, 0.0f) + log1pf(expf(-fabsf(x)));
}

// ---------------------------------------------------------------------------
// C[M,64] = A[M,K] @ B[K,64] + bias[64].   M multiple of 16, K multiple of 4,
// K <= 92.  Block = 128 threads = 4 waves; wave w owns the 16x16 tile at
// (blockIdx.x*16, w*16).  A tile (16xK) and B panel (Kx64) staged in LDS.
// ---------------------------------------------------------------------------
__global__ __launch_bounds__(128) void gemm_wmma_n64(
    const float* __restrict__ A, const float* __restrict__ B,
    const float* __restrict__ bias, float* __restrict__ C, int K) {
  __shared__ float sA[16 * 92];
  __shared__ float sB[92 * 64];
  const int tid = threadIdx.x;
  const int m_base = blockIdx.x * 16;

  for (int i = tid; i < 16 * K; i += 128) {
    int r = i / K;
    int c = i - r * K;
    sA[i] = A[(size_t)(m_base + r) * K + c];
  }
  for (int i = tid; i < K * 64; i += 128) sB[i] = B[i];
  __syncthreads();

  const int wave = tid >> 5;
  const int lane = tid & 31;
  const int m    = lane & 15;
  const int khi  = (lane >> 4) << 1;  // 0 for lanes 0-15, 2 for lanes 16-31
  const int col  = (wave << 4) + m;   // global N column of this lane

  v8f acc = {0.f, 0.f, 0.f, 0.f, 0.f, 0.f, 0.f, 0.f};
  for (int k = 0; k < K; k += 4) {
    v2f a, b;
    // A 16x4 fragment: lane m, VGPR0/1 = K = k+khi, k+khi+1
    a.x = sA[m * K + k + khi];
    a.y = sA[m * K + k + khi + 1];
    // B 4x16 fragment: row striped across lanes within a VGPR
    b.x = sB[(k + khi) * 64 + col];
    b.y = sB[(k + khi + 1) * 64 + col];
    acc = __builtin_amdgcn_wmma_f32_16x16x4_f32(false, a, false, b, (short)0,
                                                acc, false, false);
  }

  const float bv = bias[col];
  const int rbase = m_base + ((lane >> 4) << 3);  // C VGPR r -> row r (+8 hi)
#pragma unroll
  for (int r = 0; r < 8; ++r)
    C[(size_t)(rbase + r) * 64 + col] = acc[r] + bv;
}

// ---------------------------------------------------------------------------
// Degrees + reciprocals
// ---------------------------------------------------------------------------
__global__ void degrees_kernel(const int* __restrict__ node_idx,
                               const int* __restrict__ edge_idx,
                               float* __restrict__ deg_n,
                               float* __restrict__ deg_e) {
  int i = blockIdx.x * blockDim.x + threadIdx.x;
  if (i >= N_INC) return;
  atomicAdd(&deg_n[node_idx[i]], 1.0f);
  atomicAdd(&deg_e[edge_idx[i]], 1.0f);
}

__global__ void reciprocal_kernel(float* __restrict__ a, int n) {
  int i = blockIdx.x * blockDim.x + threadIdx.x;
  if (i >= n) return;
  float d = a[i];
  a[i] = (d > 0.0f) ? (1.0f / d) : 0.0f;
}

// ---------------------------------------------------------------------------
// node -> hyperedge scatter:  e_feat[e] += src[n]   (16 threads / pair, f4)
// ---------------------------------------------------------------------------
__global__ void scatter_n2e(const float* __restrict__ src,
                            const int* __restrict__ node_idx,
                            const int* __restrict__ edge_idx,
                            float* __restrict__ e_feat) {
  int t = blockIdx.x * blockDim.x + threadIdx.x;
  int pair = t >> 4;
  if (pair >= N_INC) return;
  int f4 = (t & 15) << 2;
  int n = node_idx[pair];
  int e = edge_idx[pair];
  const float4 v = *reinterpret_cast<const float4*>(src + (size_t)n * 64 + f4);
  float* dst = e_feat + (size_t)e * 64 + f4;
  atomicAdd(dst + 0, v.x);
  atomicAdd(dst + 1, v.y);
  atomicAdd(dst + 2, v.z);
  atomicAdd(dst + 3, v.w);
}

// hyperedge -> node scatter with Binv scale:  nodebuf[n] += e_feat[e]*Binv[e]
__global__ void scatter_e2n(const float* __restrict__ e_feat,
                            const float* __restrict__ Binv,
                            const int* __restrict__ node_idx,
                            const int* __restrict__ edge_idx,
                            float* __restrict__ nodebuf) {
  int t = blockIdx.x * blockDim.x + threadIdx.x;
  int pair = t >> 4;
  if (pair >= N_INC) return;
  int f4 = (t & 15) << 2;
  int n = node_idx[pair];
  int e = edge_idx[pair];
  float s = Binv[e];
  const float4 v = *reinterpret_cast<const float4*>(e_feat + (size_t)e * 64 + f4);
  float* dst = nodebuf + (size_t)n * 64 + f4;
  atomicAdd(dst + 0, v.x * s);
  atomicAdd(dst + 1, v.y * s);
  atomicAdd(dst + 2, v.z * s);
  atomicAdd(dst + 3, v.w * s);
}

// ---------------------------------------------------------------------------
// In-place Dinv scale + per-feature sum / sumsq accumulation (BN stats).
// blockDim = 64 (one column per thread; coalesced rows).
// ---------------------------------------------------------------------------
__global__ __launch_bounds__(64) void scale_stats_kernel(
    float* __restrict__ buf, const float* __restrict__ Dinv,
    float* __restrict__ sums /* [128]: sum, sumsq */, int rows_per_block) {
  int f = threadIdx.x;
  int r0 = blockIdx.x * rows_per_block;
  int r1 = r0 + rows_per_block;
  if (r1 > N_NODES) r1 = N_NODES;
  float s = 0.0f, ss = 0.0f;
  for (int r = r0; r < r1; ++r) {
    float v = buf[(size_t)r * 64 + f] * Dinv[r];
    buf[(size_t)r * 64 + f] = v;
    s += v;
    ss += v * v;
  }
  atomicAdd(&sums[f], s);
  atomicAdd(&sums[64 + f], ss);
}

// BN apply (training-mode, biased var) + softplus
__global__ void bn_apply_kernel(const float* __restrict__ buf,
                                const float* __restrict__ sums,
                                const float* __restrict__ gamma,
                                const float* __restrict__ beta,
                                float* __restrict__ h) {
  int idx = blockIdx.x * blockDim.x + threadIdx.x;
  if (idx >= N_NODES * 64) return;
  int f = idx & 63;
  const float invN = 1.0f / (float)N_NODES;
  float mean = sums[f] * invN;
  float var = sums[64 + f] * invN - mean * mean;
  float v = gamma[f] * (buf[idx] - mean) * rsqrtf(var + BN_EPS) + beta[f];
  h[idx] = softplus_f(v);
}

// ---------------------------------------------------------------------------
// scatter-mean pooling over graphs
// ---------------------------------------------------------------------------
__global__ void pool_cnt_kernel(const int* __restrict__ batch,
                                float* __restrict__ cnt) {
  int n = blockIdx.x * blockDim.x + threadIdx.x;
  if (n >= N_NODES) return;
  atomicAdd(&cnt[batch[n]], 1.0f);
}

__global__ void pool_sum_kernel(const float* __restrict__ h,
                                const int* __restrict__ batch,
                                float* __restrict__ pooled) {
  int t = blockIdx.x * blockDim.x + threadIdx.x;
  int n = t >> 4;
  if (n >= N_NODES) return;
  int f4 = (t & 15) << 2;
  int b = batch[n];
  const float4 v = *reinterpret_cast<const float4*>(h + (size_t)n * 64 + f4);
  float* dst = pooled + (size_t)b * 64 + f4;
  atomicAdd(dst + 0, v.x);
  atomicAdd(dst + 1, v.y);
  atomicAdd(dst + 2, v.z);
  atomicAdd(dst + 3, v.w);
}

// ---------------------------------------------------------------------------
// Head: g = softplus(pooled/cnt); g = softplus(g @ fcW + fcb); out = g @ fcoW
// One block of 128 threads per graph.
// ---------------------------------------------------------------------------
__global__ __launch_bounds__(128) void head_kernel(
    const float* __restrict__ pooled, const float* __restrict__ cnt,
    const float* __restrict__ fcW, const float* __restrict__ fcb,
    const float* __restrict__ fcoW, const float* __restrict__ fcob,
    float* __restrict__ out) {
  __shared__ float g[64];
  __shared__ float red[128];
  int b = blockIdx.x;
  int t = threadIdx.x;
  if (t < 64) {
    float c = fmaxf(cnt[b], 1.0f);
    g[t] = softplus_f(pooled[(size_t)b * 64 + t] / c);
  }
  __syncthreads();
  float acc = fcb[t];
#pragma unroll 8
  for (int f = 0; f < 64; ++f) acc += g[f] * fcW[f * H_DIM + t];
  red[t] = softplus_f(acc) * fcoW[t];
  __syncthreads();
  for (int s = 64; s > 0; s >>= 1) {
    if (t < s) red[t] += red[t + s];
    __syncthreads();
  }
  if (t == 0) out[b] = red[0] + fcob[0];
}

// ---------------------------------------------------------------------------
static inline size_t alignUp(size_t x) { return (x + 255) & ~(size_t)255; }

extern "C" void kernel_launch(void* const* d_in, const int* in_sizes, int n_in,
                              void* d_out, int out_size, void* d_ws,
                              size_t ws_size, hipStream_t stream) {
  const float* x      = (const float*)d_in[0];
  const float* W_emb  = (const float*)d_in[1];
  const float* b_emb  = (const float*)d_in[2];
  const float* conv_W = (const float*)d_in[3];   // [3,64,64]
  const float* conv_b = (const float*)d_in[4];   // [3,64]
  const float* bn_g   = (const float*)d_in[5];   // [3,64]
  const float* bn_b   = (const float*)d_in[6];   // [3,64]
  const float* fc_W   = (const float*)d_in[7];   // [64,128]
  const float* fc_b   = (const float*)d_in[8];   // [128]
  const float* fco_W  = (const float*)d_in[9];   // [128,1]
  const float* fco_b  = (const float*)d_in[10];  // [1]
  const int* node_idx = (const int*)d_in[11];
  const int* edge_idx = (const int*)d_in[12];
  const int* batch    = (const int*)d_in[13];
  float* out = (float*)d_out;

  // workspace carve-up
  char* ws = (char*)d_ws;
  float* h      = (float*)ws; ws += alignUp((size_t)N_NODES * 64 * 4);
  float* buf    = (float*)ws; ws += alignUp((size_t)N_NODES * 64 * 4);   // hW then nodebuf
  float* e_feat = (float*)ws; ws += alignUp((size_t)N_EDGES * 64 * 4);
  float* Binv   = (float*)ws; ws += alignUp((size_t)N_EDGES * 4);
  float* Dinv   = (float*)ws; ws += alignUp((size_t)N_NODES * 4);
  float* bnsums = (float*)ws; ws += alignUp(128 * 4);
  float* pooled = (float*)ws; ws += alignUp((size_t)N_GRAPHS * 64 * 4);
  float* cnt    = (float*)ws; ws += alignUp((size_t)N_GRAPHS * 4);

  const int B256 = 256;
  const int ROWS_PER_STAT_BLOCK = 256;
  const int statGrid = (N_NODES + ROWS_PER_STAT_BLOCK - 1) / ROWS_PER_STAT_BLOCK;
  const int incGrid = ((size_t)N_INC * 16 + B256 - 1) / B256;

  // degrees (recomputed every call; ws is not preserved between calls)
  hipMemsetAsync(Binv, 0, (size_t)N_EDGES * 4, stream);
  hipMemsetAsync(Dinv, 0, (size_t)N_NODES * 4, stream);
  degrees_kernel<<<(N_INC + B256 - 1) / B256, B256, 0, stream>>>(
      node_idx, edge_idx, Dinv, Binv);
  reciprocal_kernel<<<(N_EDGES + B256 - 1) / B256, B256, 0, stream>>>(Binv, N_EDGES);
  reciprocal_kernel<<<(N_NODES + B256 - 1) / B256, B256, 0, stream>>>(Dinv, N_NODES);

  // embedding: h = x @ W_emb + b_emb
  gemm_wmma_n64<<<N_NODES / 16, 128, 0, stream>>>(x, W_emb, b_emb, h, ATOM_DIM);

  for (int l = 0; l < NUM_LAYERS; ++l) {
    // buf = h @ conv_W[l] + conv_b[l]
    gemm_wmma_n64<<<N_NODES / 16, 128, 0, stream>>>(
        h, conv_W + (size_t)l * 64 * 64, conv_b + (size_t)l * 64, buf, NODE_DIM);

    // node -> edge mean (sum, then Binv applied on the way back out)
    hipMemsetAsync(e_feat, 0, (size_t)N_EDGES * 64 * 4, stream);
    scatter_n2e<<<incGrid, B256, 0, stream>>>(buf, node_idx, edge_idx, e_feat);

    // edge -> node (buf reused as node accumulator)
    hipMemsetAsync(buf, 0, (size_t)N_NODES * 64 * 4, stream);
    scatter_e2n<<<incGrid, B256, 0, stream>>>(e_feat, Binv, node_idx, edge_idx, buf);

    // Dinv scale (in place) + BN stats
    hipMemsetAsync(bnsums, 0, 128 * 4, stream);
    scale_stats_kernel<<<statGrid, 64, 0, stream>>>(buf, Dinv, bnsums,
                                                    ROWS_PER_STAT_BLOCK);
    // BN + softplus -> h
    bn_apply_kernel<<<(N_NODES * 64 + B256 - 1) / B256, B256, 0, stream>>>(
        buf, bnsums, bn_g + (size_t)l * 64, bn_b + (size_t)l * 64, h);
  }

  // scatter-mean pooling
  hipMemsetAsync(pooled, 0, (size_t)N_GRAPHS * 64 * 4, stream);
  hipMemsetAsync(cnt, 0, (size_t)N_GRAPHS * 4, stream);
  pool_cnt_kernel<<<(N_NODES + B256 - 1) / B256, B256, 0, stream>>>(batch, cnt);
  pool_sum_kernel<<<((size_t)N_NODES * 16 + B256 - 1) / B256, B256, 0, stream>>>(
      h, batch, pooled);

  // head MLP -> out[512]
  head_kernel<<<N_GRAPHS, 128, 0, stream>>>(pooled, cnt, fc_W, fc_b, fco_W,
                                            fco_b, out);
}